// HungarianMatcher_28089086116288
// MI455X (gfx1250) — compile-verified
//
#include <hip/hip_runtime.h>
#include <hip/hip_bf16.h>
#include <cstdint>

// Problem constants (from reference setup_inputs)
#define BS   16
#define NQ   900
#define NT   60
#define NROW (BS * NQ)   // 14400
#define NCOL (BS * NT)   // 960

typedef float v2f __attribute__((ext_vector_type(2)));
typedef float v8f __attribute__((ext_vector_type(8)));

// ---------------------------------------------------------------------------
// Kernel 1: cost matrix via V_WMMA_F32_16X16X4_F32.
// One wave (32 lanes) computes one 16x16 tile of C.
//   C[gi][gj] = -2*sigmoid(logit[gi]) + sum_k 5*|ob[gi][k] - tb[gj][k]|
// The K-sum is computed as  D = sum_{k,t} A_{k,t} x B_t + C0  where
//   A_{k,t}[m][kk] = 5*|ob[gi0+m][k] - tb[gj0+4t+kk][k]|   (16x4 slice)
//   B_t[kk][n]     = delta(n, 4t+kk)                        (4x16 identity slice)
// so D[m][4t+kk] accumulates the abs-diff for feature k. C0 carries the logit
// term (depends only on the row). 20 WMMAs per tile; kernel is store-bound.
// Also scatters the batch-diagonal entries transposed into blockT[b][t][q]
// for the Hungarian solver.
// ---------------------------------------------------------------------------
__global__ __launch_bounds__(128) void cost_kernel(
    const float* __restrict__ out_boxes,   // [BS,NQ,4]
    const float* __restrict__ out_angle,   // [BS,NQ,1]
    const float* __restrict__ out_logits,  // [BS,NQ,1]
    const float* __restrict__ tgt_boxes,   // [BS,NT,4]
    const float* __restrict__ tgt_angle,   // [BS,NT,1]
    float* __restrict__ C,                 // [NROW,NCOL]
    float* __restrict__ blockT)            // [BS,NT,NQ]
{
    __shared__ float tbs[16][5];           // target tile (shared by all 4 waves)

    const int tid = threadIdx.y * 32 + threadIdx.x;
    const int gj0 = blockIdx.x * 16;       // column tile base (0..944)

    if (tid < 80) {                        // 16 targets x 5 features
        const int c = tid / 5, k = tid % 5;
        const int gj = gj0 + c;
        const int jb = gj / NT, tt = gj % NT;
        const int tflat = jb * NT + tt;
        tbs[c][k] = (k < 4) ? tgt_boxes[tflat * 4 + k] : tgt_angle[tflat];
    }
    __syncthreads();

    const int lane = threadIdx.x;          // 0..31
    const int half = lane >> 4;            // 0 or 1
    const int l    = lane & 15;
    const int rowTile = blockIdx.y * 4 + threadIdx.y;  // 0..899
    const int gi0 = rowTile * 16;
    const int gi_l = gi0 + l;

    // Per-lane query row (A-matrix row m = l for both half-waves).
    float ob[5];
    {
        const float4 b4 = *(const float4*)(out_boxes + (size_t)gi_l * 4);
        ob[0] = b4.x; ob[1] = b4.y; ob[2] = b4.z; ob[3] = b4.w;
        ob[4] = out_angle[gi_l];
    }

    // Accumulator init with the logit term. C/D layout (ISA 7.12.2):
    // VGPR r: lanes 0-15 -> (M=r, N=l), lanes 16-31 -> (M=r+8, N=l).
    v8f acc = {};
    #pragma unroll
    for (int r = 0; r < 8; ++r) {
        const float x = out_logits[gi0 + half * 8 + r];
        const float prob = 1.0f / (1.0f + __expf(-x));
        acc[r] = -2.0f * prob;             // W_LOGITS * (-sigmoid)
    }

    // 20 x V_WMMA_F32_16X16X4_F32.
    // A layout: lanes 0-15 hold (m=l, K=0 in v0 / K=1 in v1);
    //           lanes 16-31 hold (m=l, K=2 / K=3)  -> kk = 2*half + {0,1}.
    // B layout mirrors: lane holds (K=2*half+{0,1}, n=l).
    #pragma unroll
    for (int t = 0; t < 4; ++t) {
        v2f bm;
        bm.x = (l == (4 * t + 2 * half))     ? 1.0f : 0.0f;
        bm.y = (l == (4 * t + 2 * half + 1)) ? 1.0f : 0.0f;
        #pragma unroll
        for (int k = 0; k < 5; ++k) {
            const float obk = ob[k];
            v2f am;
            am.x = 5.0f * fabsf(obk - tbs[4 * t + 2 * half    ][k]);
            am.y = 5.0f * fabsf(obk - tbs[4 * t + 2 * half + 1][k]);
            acc = __builtin_amdgcn_wmma_f32_16x16x4_f32(
                /*neg_a=*/false, am, /*neg_b=*/false, bm,
                /*c_mod=*/(short)0, acc, /*reuse_a=*/false, /*reuse_b=*/false);
        }
    }

    // Store tile; scatter diagonal-block entries transposed for the LSA stage.
    const int col = gj0 + l;
    const int jb = col / NT, tt = col % NT;
    #pragma unroll
    for (int r = 0; r < 8; ++r) {
        const int row = gi0 + half * 8 + r;
        const float val = acc[r];
        C[(size_t)row * NCOL + col] = val;
        const int bb = row / NQ;
        if (bb == jb) {
            const int q = row - bb * NQ;
            blockT[((size_t)bb * NT + tt) * NQ + q] = val;
        }
    }
}

// ---------------------------------------------------------------------------
// Kernel 2: Hungarian (e-maxx potentials), one wave32 per batch element.
// cost[t][q] = blockT[b][t][q], n=NT rows, m=NQ cols. The 901-wide inner
// scan is lane-strided; the argmin (first-index tie-break, matching
// jnp.argmin) is a 5-step __shfl_xor reduction. State lives in LDS.
// ---------------------------------------------------------------------------
__global__ __launch_bounds__(32) void hungarian_kernel(
    const float* __restrict__ blockT,      // [BS,NT,NQ]
    int* __restrict__ qoft)                // [BS,NT]
{
    const int b = blockIdx.x;
    const int lane = threadIdx.x;

    __shared__ float v[NQ + 1];
    __shared__ float minv[NQ + 1];
    __shared__ float u[NT + 1];
    __shared__ int   way[NQ + 1];
    __shared__ int   p[NQ + 1];
    __shared__ int   used[NQ + 1];

    for (int j = lane; j <= NQ; j += 32) { v[j] = 0.0f; p[j] = 0; }
    for (int i = lane; i <= NT; i += 32) u[i] = 0.0f;
    __syncthreads();

    const float BIG = 1000000000.0f;
    const float* base = blockT + (size_t)b * NT * NQ;

    for (int i = 1; i <= NT; ++i) {
        if (lane == 0) p[0] = i;
        for (int j = lane; j <= NQ; j += 32) { minv[j] = BIG; used[j] = 0; way[j] = 0; }
        __syncthreads();

        int j0 = 0;
        while (true) {
            if (lane == 0) used[j0] = 1;
            __syncthreads();

            const int i0 = p[j0];
            const float uu = u[i0];
            const float* arow = base + (size_t)(i0 - 1) * NQ;
            __builtin_prefetch(arow, 0, 1);   // global_prefetch_b8

            float bestv = BIG;
            int   bestj = NQ + 1;
            for (int j = lane; j <= NQ; j += 32) {
                if (!used[j]) {
                    const float cur = (j ? arow[j - 1] : 0.0f) - uu - v[j];
                    float mv = minv[j];
                    if (cur < mv) { mv = cur; minv[j] = cur; way[j] = j0; }
                    if (mv < bestv || (mv == bestv && j < bestj)) { bestv = mv; bestj = j; }
                }
            }
            #pragma unroll
            for (int off = 16; off > 0; off >>= 1) {
                const float ov = __shfl_xor(bestv, off, 32);
                const int   oj = __shfl_xor(bestj, off, 32);
                if (ov < bestv || (ov == bestv && oj < bestj)) { bestv = ov; bestj = oj; }
            }
            const float delta = bestv;
            const int j1 = bestj;

            for (int j = lane; j <= NQ; j += 32) {
                if (used[j]) { v[j] -= delta; u[p[j]] += delta; }  // distinct u slots
                else         { minv[j] -= delta; }
            }
            j0 = j1;
            __syncthreads();
            if (p[j0] == 0) break;
        }

        if (lane == 0) {   // augmenting-path flip (tiny, serial)
            int jj = j0;
            while (jj) { const int jn = way[jj]; p[jj] = p[jn]; jj = jn; }
        }
        __syncthreads();
    }

    // col_of_row: for each assigned column j, row p[j] got query j-1.
    for (int j = 1 + lane; j <= NQ; j += 32) {
        const int r = p[j];
        if (r) qoft[b * NT + (r - 1)] = j - 1;
    }
}

// ---------------------------------------------------------------------------
// Kernel 3: per-batch rank (values distinct -> rank == argsort position).
// index_i[b][rank] = q_of_t[t]; index_j[b][rank] = t. Written as float into
// the flattened tuple output buffer.
// ---------------------------------------------------------------------------
__global__ __launch_bounds__(64) void argsort_kernel(
    const int* __restrict__ qoft, float* __restrict__ out)
{
    const int b = blockIdx.x;
    const int t = threadIdx.x;
    if (t < NT) {
        const int q = qoft[b * NT + t];
        int rank = 0;
        for (int s = 0; s < NT; ++s) rank += (qoft[b * NT + s] < q) ? 1 : 0;
        out[b * NT + rank]            = (float)q;   // index_i
        out[BS * NT + b * NT + rank]  = (float)t;   // index_j
    }
}

// ---------------------------------------------------------------------------
extern "C" void kernel_launch(void* const* d_in, const int* in_sizes, int n_in,
                              void* d_out, int out_size, void* d_ws, size_t ws_size,
                              hipStream_t stream) {
    (void)in_sizes; (void)n_in; (void)out_size; (void)ws_size;

    const float* out_boxes  = (const float*)d_in[0];
    const float* out_angle  = (const float*)d_in[1];
    const float* out_logits = (const float*)d_in[2];
    const float* tgt_boxes  = (const float*)d_in[3];
    const float* tgt_angle  = (const float*)d_in[4];

    float* out = (float*)d_out;
    float* C = out + 2 * BS * NT;                           // after index_i, index_j
    float* blockT = (float*)d_ws;                           // [BS,NT,NQ] f32 (3.46 MB)
    int* qoft = (int*)((char*)d_ws + (size_t)BS * NT * NQ * sizeof(float));

    dim3 blk(32, 4);                 // 4 waves, one 16x16 tile each (stacked rows)
    dim3 grd(NCOL / 16, NROW / 16 / 4);   // (60, 225)
    cost_kernel<<<grd, blk, 0, stream>>>(out_boxes, out_angle, out_logits,
                                         tgt_boxes, tgt_angle, C, blockT);
    hungarian_kernel<<<BS, 32, 0, stream>>>(blockT, qoft);
    argsort_kernel<<<BS, 64, 0, stream>>>(qoft, out);
}